// GCN_83502754169281
// MI455X (gfx1250) — compile-verified
//
#include <hip/hip_runtime.h>
#include <hip/hip_bf16.h>

typedef __attribute__((ext_vector_type(2))) float v2f;
typedef __attribute__((ext_vector_type(8))) float v8f;

#define TN      80000
#define FIN     128
#define HCH     128
#define NEDGE   1280000
#define BGRAPH  8
#define NPG     10000
#define NG      8000
#define LN_EPS  1e-5f

// ---------------------------------------------------------------- zero agg
__global__ void k_zero(float* __restrict__ p, int n4) {
    int i = blockIdx.x * blockDim.x + threadIdx.x;
    if (i < n4) ((float4*)p)[i] = make_float4(0.f, 0.f, 0.f, 0.f);
}

// ------------------------------------------------- u = colmean(W_rel2), etc.
__global__ void k_uvs(const float* __restrict__ Wrel2, const float* __restrict__ Wroot2,
                      const float* __restrict__ b2,
                      float* __restrict__ u, float* __restrict__ v, float* __restrict__ s) {
    int k = threadIdx.x;                    // 0..127
    float su = 0.f, sv = 0.f;
    for (int n = 0; n < HCH; ++n) {
        su += Wrel2[n * HCH + k];
        sv += Wroot2[n * HCH + k];
    }
    u[k] = su * (1.f / HCH);
    v[k] = sv * (1.f / HCH);
    if (k == 0) {
        float sb = 0.f;
        for (int n = 0; n < HCH; ++n) sb += b2[n];
        *s = sb * (1.f / HCH);
    }
}

// --------------------------------- layer-1 message scatter: agg[dst] += w*x[src]
// one wave per edge: lane*4 channels -> coalesced 512B gather + float4 of atomics
__global__ void k_scatter_vec(const float* __restrict__ x,
                              const int* __restrict__ src, const int* __restrict__ dst,
                              const float* __restrict__ w, float* __restrict__ agg) {
    long long gid = (long long)blockIdx.x * blockDim.x + threadIdx.x;
    int e = (int)(gid >> 5);
    if (e >= NEDGE) return;
    int c = ((int)gid & 31) * 4;
    int sI = src[e], dI = dst[e];
    float we = w[e];
    float4 xv = *(const float4*)(x + (size_t)sI * FIN + c);
    float* a = agg + (size_t)dI * FIN + c;
    atomicAdd(a + 0, we * xv.x);
    atomicAdd(a + 1, we * xv.y);
    atomicAdd(a + 2, we * xv.z);
    atomicAdd(a + 3, we * xv.w);
}

// ------------------- fused: h = relu([agg|x] @ [Wrel1|Wroot1]^T + b1) (WMMA f32)
//                     then  t = h.u ; pooled = h.v + s   (h never materialized)
// block = 4 waves, each wave owns 16 rows x all 128 cols. TN % 64 == 0 -> EXEC all-1s.
__global__ __launch_bounds__(128) void k_gemm_fused(
        const float* __restrict__ agg, const float* __restrict__ x,
        const float* __restrict__ Wrel, const float* __restrict__ b1,
        const float* __restrict__ Wroot,
        const float* __restrict__ u, const float* __restrict__ v,
        const float* __restrict__ s,
        float* __restrict__ t, float* __restrict__ pooled) {
    const int lane  = threadIdx.x & 31;
    const int wv    = threadIdx.x >> 5;
    const int row0  = blockIdx.x * 64 + wv * 16;
    const int m     = lane & 15;          // A-row / B-col index within tile
    const int khalf = (lane >> 4) * 2;    // K sub-offset per ISA 16x4 f32 layout

    v8f acc[8];
#pragma unroll
    for (int n = 0; n < 8; ++n) acc[n] = (v8f){0.f,0.f,0.f,0.f,0.f,0.f,0.f,0.f};

    // K-loop 1: agg @ Wrel^T   (D[m,n] = sum_k agg[m,k] * Wrel[n,k])
    const float* Arow = agg + (size_t)(row0 + m) * HCH;
#pragma unroll 2
    for (int k0 = 0; k0 < HCH; k0 += 4) {
        v2f a = *(const v2f*)(Arow + k0 + khalf);
#pragma unroll
        for (int n = 0; n < 8; ++n) {
            v2f b = *(const v2f*)(Wrel + (size_t)(n * 16 + m) * HCH + k0 + khalf);
            acc[n] = __builtin_amdgcn_wmma_f32_16x16x4_f32(
                false, a, false, b, (short)0, acc[n], false, false);
        }
    }
    // K-loop 2: x @ Wroot^T
    const float* Xrow = x + (size_t)(row0 + m) * FIN;
#pragma unroll 2
    for (int k0 = 0; k0 < FIN; k0 += 4) {
        v2f a = *(const v2f*)(Xrow + k0 + khalf);
#pragma unroll
        for (int n = 0; n < 8; ++n) {
            v2f b = *(const v2f*)(Wroot + (size_t)(n * 16 + m) * FIN + k0 + khalf);
            acc[n] = __builtin_amdgcn_wmma_f32_16x16x4_f32(
                false, a, false, b, (short)0, acc[n], false, false);
        }
    }

    // epilogue: bias + relu + dual dot products (u for t, v for pooled)
    float tacc[8], pacc[8];
#pragma unroll
    for (int i = 0; i < 8; ++i) { tacc[i] = 0.f; pacc[i] = 0.f; }
#pragma unroll
    for (int n = 0; n < 8; ++n) {
        int col = n * 16 + m;             // D-layout: this lane holds column (lane&15)
        float bias = b1[col], uu = u[col], vv = v[col];
#pragma unroll
        for (int i = 0; i < 8; ++i) {     // rows i + 8*(lane>>4)
            float hv = fmaxf(acc[n][i] + bias, 0.f);
            tacc[i] = fmaf(hv, uu, tacc[i]);
            pacc[i] = fmaf(hv, vv, pacc[i]);
        }
    }
    // reduce across the 16 lanes that share the same rows
#pragma unroll
    for (int off = 1; off < 16; off <<= 1) {
#pragma unroll
        for (int i = 0; i < 8; ++i) {
            tacc[i] += __shfl_xor(tacc[i], off, 16);
            pacc[i] += __shfl_xor(pacc[i], off, 16);
        }
    }
    if (m == 0) {
        float sb = *s;
        int rbase = row0 + (lane >> 4) * 8;
#pragma unroll
        for (int i = 0; i < 8; ++i) {
            t[rbase + i]      = tacc[i];
            pooled[rbase + i] = pacc[i] + sb;
        }
    }
}

// ---------------- layer-2 collapsed scatter: pooled[dst] += w * t[src] (scalar!)
__global__ void k_scatter_scalar(const float* __restrict__ t,
                                 const int* __restrict__ src, const int* __restrict__ dst,
                                 const float* __restrict__ w, float* __restrict__ pooled) {
    int e = blockIdx.x * blockDim.x + threadIdx.x;
    if (e >= NEDGE) return;
    atomicAdd(&pooled[dst[e]], w[e] * t[src[e]]);
}

// ------------------------- per-graph LayerNorm(8000) + log_softmax, 1 block/graph
__global__ __launch_bounds__(1024) void k_norm_softmax(
        const float* __restrict__ pooled, const float* __restrict__ gamma,
        const float* __restrict__ beta, float* __restrict__ out) {
    __shared__ float sy[NG];
    __shared__ float red[1024];
    const int b = blockIdx.x, tid = threadIdx.x;
    const float* g = pooled + (size_t)b * NPG;

    float lsum = 0.f, lsq = 0.f;
    for (int j = tid; j < NG; j += 1024) {
        float val = g[j];
        sy[j] = val;
        lsum += val;
        lsq = fmaf(val, val, lsq);
    }
    red[tid] = lsum; __syncthreads();
    for (int st = 512; st > 0; st >>= 1) { if (tid < st) red[tid] += red[tid + st]; __syncthreads(); }
    float mu = red[0] * (1.f / NG); __syncthreads();
    red[tid] = lsq; __syncthreads();
    for (int st = 512; st > 0; st >>= 1) { if (tid < st) red[tid] += red[tid + st]; __syncthreads(); }
    float var = red[0] * (1.f / NG) - mu * mu;
    float rs  = rsqrtf(var + LN_EPS);
    __syncthreads();

    float lmax = -3.4e38f;
    for (int j = tid; j < NG; j += 1024) {
        float y = (sy[j] - mu) * rs * gamma[j] + beta[j];
        sy[j] = y;
        lmax = fmaxf(lmax, y);
    }
    red[tid] = lmax; __syncthreads();
    for (int st = 512; st > 0; st >>= 1) { if (tid < st) red[tid] = fmaxf(red[tid], red[tid + st]); __syncthreads(); }
    float mx = red[0]; __syncthreads();

    float lexp = 0.f;
    for (int j = tid; j < NG; j += 1024) lexp += expf(sy[j] - mx);
    red[tid] = lexp; __syncthreads();
    for (int st = 512; st > 0; st >>= 1) { if (tid < st) red[tid] += red[tid + st]; __syncthreads(); }
    float logZ = logf(red[0]) + mx;
    __syncthreads();

    for (int j = tid; j < NG; j += 1024)
        out[(size_t)b * NG + j] = sy[j] - logZ;
}

// ------------------------------------------------------------------ launcher
extern "C" void kernel_launch(void* const* d_in, const int* in_sizes, int n_in,
                              void* d_out, int out_size, void* d_ws, size_t ws_size,
                              hipStream_t stream) {
    const float* x       = (const float*)d_in[0];
    const int*   eidx    = (const int*)d_in[1];   // [2, E]; JAX default x64-off -> int32
    // d_in[2] = batch (unused: layout is implicit in the reshape)
    const float* ew      = (const float*)d_in[3];
    const float* Wrel1   = (const float*)d_in[4];
    const float* brel1   = (const float*)d_in[5];
    const float* Wroot1  = (const float*)d_in[6];
    const float* Wrel2   = (const float*)d_in[7];
    const float* brel2   = (const float*)d_in[8];
    const float* Wroot2  = (const float*)d_in[9];
    const float* gamma   = (const float*)d_in[10];
    const float* beta    = (const float*)d_in[11];
    const int* src = eidx;
    const int* dst = eidx + NEDGE;

    // workspace layout (all offsets 256B-aligned by construction)
    char*  ws     = (char*)d_ws;
    float* agg    = (float*)ws;                                   // TN*H  (41 MB)
    float* t      = (float*)(ws + (size_t)TN * HCH * 4);          // TN
    float* pooled = t + TN;                                       // TN
    float* u      = pooled + TN;                                  // H
    float* v      = u + HCH;                                      // H
    float* s      = v + HCH;                                      // 1

    int n4 = TN * HCH / 4;
    k_zero<<<(n4 + 255) / 256, 256, 0, stream>>>(agg, n4);
    k_uvs<<<1, HCH, 0, stream>>>(Wrel2, Wroot2, brel2, u, v, s);

    long long sthreads = (long long)NEDGE * 32;
    k_scatter_vec<<<(int)(sthreads / 256), 256, 0, stream>>>(x, src, dst, ew, agg);

    k_gemm_fused<<<TN / 64, 128, 0, stream>>>(agg, x, Wrel1, brel1, Wroot1,
                                              u, v, s, t, pooled);

    k_scatter_scalar<<<NEDGE / 256, 256, 0, stream>>>(t, src, dst, ew, pooled);

    k_norm_softmax<<<BGRAPH, 1024, 0, stream>>>(pooled, gamma, beta, (float*)d_out);
}